// dbi_focal_3D_22256520528038
// MI455X (gfx1250) — compile-verified
//
#include <hip/hip_runtime.h>
#include <hip/hip_bf16.h>

// Problem constants (match reference)
constexpr int B = 2, C = 4, D = 64, H = 192, W = 192;
constexpr int S  = H * W;          // 36864 voxels per slice
constexpr int NS = B * D;          // 128 slices
constexpr int NV = NS * S;         // 4,718,592 voxels
constexpr float FINF = 1.0e12f;    // reference INF
constexpr int NPART = NV / 256;    // 18432 partial sums (grid covers NV exactly)

typedef __attribute__((ext_vector_type(2))) float v2f;
typedef __attribute__((ext_vector_type(8))) float v8f;

#if defined(__gfx1250__) && \
    __has_builtin(__builtin_amdgcn_global_load_async_to_lds_b32) && \
    __has_builtin(__builtin_amdgcn_s_wait_asynccnt)
#define HAS_ASYNC_LDS 1
#else
#define HAS_ASYNC_LDS 0
#endif

#if defined(__gfx1250__) && __has_builtin(__builtin_amdgcn_wmma_f32_16x16x4_f32)
#define HAS_WMMA_F32 1
#else
#define HAS_WMMA_F32 0
#endif

// ---------------------------------------------------------------------------
// K0: zero the weight map accumulator (ws is poisoned 0xAA by the harness)
// ---------------------------------------------------------------------------
__global__ void k_zero(float* __restrict__ wmap) {
    int v = blockIdx.x * blockDim.x + threadIdx.x;
    if (v < NV) wmap[v] = 0.0f;
}

// ---------------------------------------------------------------------------
// K1: per-voxel log-softmax gather, pt, argmax; compress target to u8
// ---------------------------------------------------------------------------
__global__ void k_logits(const float* __restrict__ x, const int* __restrict__ tgt,
                         float* __restrict__ logpt,
                         unsigned char* __restrict__ pred8,
                         unsigned char* __restrict__ tgt8) {
    int v = blockIdx.x * blockDim.x + threadIdx.x;
    if (v >= NV) return;
    int s = v / S, p = v - s * S;
    int b = s / D, d = s - b * D;
    size_t base = ((size_t)(b * C * D + d)) * S + p;   // channel 0
    size_t cs   = (size_t)D * S;                       // channel stride
    float x0 = x[base], x1 = x[base + cs], x2 = x[base + 2 * cs], x3 = x[base + 3 * cs];
    float mx = fmaxf(fmaxf(x0, x1), fmaxf(x2, x3));
    float e0 = expf(x0 - mx), e1 = expf(x1 - mx), e2 = expf(x2 - mx), e3 = expf(x3 - mx);
    float lse = mx + logf(e0 + e1 + e2 + e3);
    int am = 0; float bv = x0;                          // first max on ties
    if (x1 > bv) { bv = x1; am = 1; }
    if (x2 > bv) { bv = x2; am = 2; }
    if (x3 > bv) { bv = x3; am = 3; }
    int t = tgt[v];
    float xt = (t == 0) ? x0 : (t == 1) ? x1 : (t == 2) ? x2 : x3;
    logpt[v] = xt - lse;
    pred8[v] = (unsigned char)am;
    tgt8[v]  = (unsigned char)t;
}

// ---------------------------------------------------------------------------
// K2: per-slice class presence (c=1..3) and "empty" (all zero) flags
// ---------------------------------------------------------------------------
__global__ void k_stats(const unsigned char* __restrict__ tgt8,
                        int* __restrict__ present /*3*NS*/, int* __restrict__ empty /*NS*/) {
    int s = blockIdx.x;
    __shared__ int fl[3];
    if (threadIdx.x < 3) fl[threadIdx.x] = 0;
    __syncthreads();
    int a1 = 0, a2 = 0, a3 = 0;
    const unsigned char* T = tgt8 + (size_t)s * S;
    for (int p = threadIdx.x; p < S; p += blockDim.x) {
        int t = T[p];
        a1 |= (t == 1); a2 |= (t == 2); a3 |= (t == 3);
    }
    if (a1) atomicOr(&fl[0], 1);
    if (a2) atomicOr(&fl[1], 1);
    if (a3) atomicOr(&fl[2], 1);
    __syncthreads();
    if (threadIdx.x == 0) {
        present[0 * NS + s] = fl[0];
        present[1 * NS + s] = fl[1];
        present[2 * NS + s] = fl[2];
        empty[s] = !(fl[0] | fl[1] | fl[2]);
    }
}

// ---------------------------------------------------------------------------
// K3: EDT pass 1 along W (two O(W) scans per row; seeds = boundary of tgt==c)
// ---------------------------------------------------------------------------
__global__ void k_edt_rows(const unsigned char* __restrict__ tgt8,
                           float* __restrict__ g, int c) {
    int s = blockIdx.x;
    int h = threadIdx.x;                      // 0..H-1
    const unsigned char* T  = tgt8 + (size_t)s * S;
    const unsigned char* Rm = T + (size_t)h * W;
    const unsigned char* Ru = (h > 0)     ? Rm - W : nullptr;
    const unsigned char* Rd = (h < H - 1) ? Rm + W : nullptr;
    float* G = g + (size_t)s * S + (size_t)h * W;

    unsigned int seed[6] = {0u, 0u, 0u, 0u, 0u, 0u};
    bool mPrev = false;
    bool mCur  = (Rm[0] == c);
    float dl = 1.0e6f;                        // dl^2 >= FINF when no seed yet
    for (int w = 0; w < W; ++w) {
        bool mNext = (w + 1 < W) && (Rm[w + 1] == c);
        bool up = Ru && (Ru[w] == c);
        bool dn = Rd && (Rd[w] == c);
        bool bnd = mCur && !(up && dn && mPrev && mNext);
        if (bnd) { seed[w >> 5] |= (1u << (w & 31)); dl = 0.0f; }
        else     { dl += 1.0f; }
        G[w] = dl * dl;                       // left-scan squared distance
        mPrev = mCur; mCur = mNext;
    }
    float dr = 1.0e6f;
    for (int w = W - 1; w >= 0; --w) {
        if (seed[w >> 5] & (1u << (w & 31))) dr = 0.0f; else dr += 1.0f;
        float v = fminf(G[w], dr * dr);
        G[w] = fminf(v, FINF);
    }
}

// ---------------------------------------------------------------------------
// K4: EDT pass 2 along H (exact min-plus). Column staged in LDS via the
// CDNA5 async-to-LDS path (ASYNCcnt-tracked) when available; fused with
// error/presence gating and wmap accumulation. Last class also applies the
// empty-slice +10 term (must land before the min/max normalization).
// ---------------------------------------------------------------------------
__global__ void k_edt_cols(const float* __restrict__ g,
                           const unsigned char* __restrict__ tgt8,
                           const unsigned char* __restrict__ pred8,
                           const int* __restrict__ present,
                           const int* __restrict__ empty,
                           float* __restrict__ wmap, int c, int lastClass) {
    int w = blockIdx.x;          // 0..W-1
    int s = blockIdx.y;          // 0..NS-1
    int h = threadIdx.x;         // 0..H-1
    __shared__ float gcol[H];
    size_t base = (size_t)s * S;
    int p = h * W + w;
    // prefetch the gating bytes needed after the min-plus loop (global_prefetch_b8)
    __builtin_prefetch(tgt8 + base + p, 0, 1);
    __builtin_prefetch(pred8 + base + p, 0, 1);

#if HAS_ASYNC_LDS
    // CDNA5 async copy: per-lane GLOBAL_LOAD_ASYNC_TO_LDS_B32, no VGPR bounce.
    // Builtin signature (from probe): (as(1) int*, as(3) int*, imm offset, imm cpol)
    {
        void* srcnc = const_cast<void*>(
            (const void*)(g + base + (size_t)h * W + w));
        __builtin_amdgcn_global_load_async_to_lds_b32(
            (__attribute__((address_space(1))) int*)srcnc,
            (__attribute__((address_space(3))) int*)&gcol[h], 0, 0);
        __builtin_amdgcn_s_wait_asynccnt(0);
    }
#else
    gcol[h] = g[base + (size_t)h * W + w];
#endif
    __syncthreads();

    float fh = (float)h;
    float best = FINF;
#pragma unroll 8
    for (int k = 0; k < H; ++k) {
        float dh = fh - (float)k;
        best = fminf(best, __builtin_fmaf(dh, dh, gcol[k]));   // v_fma + v_min
    }
    float dist = sqrtf(fminf(best, FINF));

    int t  = tgt8[base + p];
    int pr = pred8[base + p];
    bool err = (t == c) != (pr == c);
    float add = (err && present[(c - 1) * NS + s]) ? dist : 0.0f;
    if (lastClass && empty[s] && pr != 0) add += 10.0f;
    wmap[base + p] += add;
}

// ---------------------------------------------------------------------------
// K5: per-slice min/max of wmap
// ---------------------------------------------------------------------------
__global__ void k_minmax(const float* __restrict__ wmap,
                         float* __restrict__ mn, float* __restrict__ mx) {
    int s = blockIdx.x, t = threadIdx.x;
    const float* Wp = wmap + (size_t)s * S;
    float lmn = 3.4e38f, lmx = -3.4e38f;
    for (int p = t; p < S; p += blockDim.x) {
        __builtin_prefetch(Wp + p + blockDim.x, 0, 1);
        float v = Wp[p];
        lmn = fminf(lmn, v); lmx = fmaxf(lmx, v);
    }
    __shared__ float smn[256], smx[256];
    smn[t] = lmn; smx[t] = lmx;
    __syncthreads();
    for (int o = 128; o > 0; o >>= 1) {
        if (t < o) { smn[t] = fminf(smn[t], smn[t + o]); smx[t] = fmaxf(smx[t], smx[t + o]); }
        __syncthreads();
    }
    if (t == 0) { mn[s] = smn[0]; mx[s] = smx[0]; }
}

// ---------------------------------------------------------------------------
// K6a: per-voxel focal loss. Per-wave sum done with V_WMMA_F32_16X16X4_F32:
// D = A(16x4, our 32 contributions + zeros) x B(4x16, all ones) + 0, so
// sum over column 0 of D = sum of all A elements = wave total. Column 0 of
// the 16x16 f32 D lives in the 8 D-VGPRs of lanes 0 and 16 (doc layout).
// Grid covers NV exactly -> EXEC is all 1s at the WMMA (ISA requirement).
// ---------------------------------------------------------------------------
__global__ void k_loss_partial(const float* __restrict__ logpt,
                               const float* __restrict__ wmap,
                               const float* __restrict__ mn,
                               const float* __restrict__ mx,
                               float* __restrict__ partial) {
    int v = blockIdx.x * 256 + threadIdx.x;   // exact cover, no masking
    int s = v / S;
    float lp = logpt[v];
    float pt = expf(lp);
    float wn = (wmap[v] - mn[s]) / (mx[s] - mn[s] + 1e-6f);
    float wei = expf(wn);
    float om = 1.0f - pt;
    float contrib = -(om * om) * lp * wei;    // GAMMA = 2

#if HAS_WMMA_F32
    v2f a; a.x = contrib; a.y = 0.0f;         // 32 contributions + 32 zeros in A
    v2f bm; bm.x = 1.0f; bm.y = 1.0f;         // B = all ones
    v8f cz = {};
    v8f dm = __builtin_amdgcn_wmma_f32_16x16x4_f32(
        /*neg_a=*/false, a, /*neg_b=*/false, bm,
        /*c_mod=*/(short)0, cz, /*reuse_a=*/false, /*reuse_b=*/false);
    float colsum = dm[0] + dm[1] + dm[2] + dm[3] + dm[4] + dm[5] + dm[6] + dm[7];
    __shared__ float sb[16];                   // 8 waves x 2 half-column sums
    int lane = threadIdx.x & 31;
    int wv   = threadIdx.x >> 5;
    if ((lane & 15) == 0) sb[wv * 2 + (lane >> 4)] = colsum;
    __syncthreads();
    if (threadIdx.x == 0) {
        float tot = 0.0f;
        for (int i = 0; i < 16; ++i) tot += sb[i];
        partial[blockIdx.x] = tot;
    }
#else
    __shared__ float sb[256];
    sb[threadIdx.x] = contrib;
    __syncthreads();
    for (int o = 128; o > 0; o >>= 1) {
        if (threadIdx.x < o) sb[threadIdx.x] += sb[threadIdx.x + o];
        __syncthreads();
    }
    if (threadIdx.x == 0) partial[blockIdx.x] = sb[0];
#endif
}

// K6b: deterministic final reduction of the partial sums
__global__ void k_sum(const float* __restrict__ partial, int n, float* __restrict__ out) {
    float a = 0.0f;
    for (int i = threadIdx.x; i < n; i += 256) a += partial[i];
    __shared__ float sb[256];
    sb[threadIdx.x] = a;
    __syncthreads();
    for (int o = 128; o > 0; o >>= 1) {
        if (threadIdx.x < o) sb[threadIdx.x] += sb[threadIdx.x + o];
        __syncthreads();
    }
    if (threadIdx.x == 0) out[0] = sb[0];
}

// ---------------------------------------------------------------------------
extern "C" void kernel_launch(void* const* d_in, const int* in_sizes, int n_in,
                              void* d_out, int out_size, void* d_ws, size_t ws_size,
                              hipStream_t stream) {
    const float* x  = (const float*)d_in[0];   // (B,C,D,H,W) f32
    const int* tgt  = (const int*)d_in[1];     // (B,1,D,H,W) i32
    float* out = (float*)d_out;

    char* ws = (char*)d_ws;
    size_t off = 0;
    auto alloc = [&](size_t bytes) -> void* {
        off = (off + 255) & ~(size_t)255;
        void* p = ws + off;
        off += bytes;
        return p;
    };
    float* logpt          = (float*)alloc((size_t)NV * 4);
    float* wmap           = (float*)alloc((size_t)NV * 4);
    float* gbuf           = (float*)alloc((size_t)NV * 4);
    unsigned char* pred8  = (unsigned char*)alloc((size_t)NV);
    unsigned char* tgt8   = (unsigned char*)alloc((size_t)NV);
    int* present          = (int*)alloc(3 * NS * 4);
    int* empty            = (int*)alloc(NS * 4);
    float* mn             = (float*)alloc(NS * 4);
    float* mx             = (float*)alloc(NS * 4);
    float* partial        = (float*)alloc((size_t)NPART * 4);
    (void)ws_size; (void)in_sizes; (void)n_in; (void)out_size;

    k_zero<<<NV / 256, 256, 0, stream>>>(wmap);
    k_logits<<<NV / 256, 256, 0, stream>>>(x, tgt, logpt, pred8, tgt8);
    k_stats<<<NS, 256, 0, stream>>>(tgt8, present, empty);

    for (int c = 1; c < 4; ++c) {
        k_edt_rows<<<NS, H, 0, stream>>>(tgt8, gbuf, c);
        dim3 g2(W, NS);
        k_edt_cols<<<g2, H, 0, stream>>>(gbuf, tgt8, pred8, present, empty, wmap, c, c == 3 ? 1 : 0);
    }

    k_minmax<<<NS, 256, 0, stream>>>(wmap, mn, mx);
    k_loss_partial<<<NPART, 256, 0, stream>>>(logpt, wmap, mn, mx, partial);
    k_sum<<<1, 256, 0, stream>>>(partial, NPART, out);
}